// WarpTPS_48387101557410
// MI455X (gfx1250) — compile-verified
//
#include <hip/hip_runtime.h>
#include <math.h>

// Problem constants from the reference
constexpr int N_ = 8, C_ = 3, H_ = 512, W_ = 512, T_ = 32;
constexpr int NS = T_ + 3;            // 35 unknowns
constexpr int NCOLS = NS + 2;         // augmented with 2 RHS columns
constexpr float EPS_ = 1e-6f;
constexpr float LN2_ = 0.6931471805599453f;

typedef __attribute__((ext_vector_type(2))) float v2f;
typedef __attribute__((ext_vector_type(4))) float v4f;
typedef __attribute__((ext_vector_type(8))) float v8f;

// U(r^2) = r^2 * log(sqrt(r^2) + eps). Branchless: at r2==0 the product is 0
// regardless of the (finite) log value, matching the reference's guarded form.
// Raw HW transcendentals: v_sqrt_f32 + v_log_f32 (log2 -> *ln2).
__device__ __forceinline__ float tpsU(float dx, float dy) {
    float r2 = dx * dx + dy * dy;
    float r  = __builtin_amdgcn_sqrtf(r2);
    float l2 = __builtin_amdgcn_logf(r + EPS_);
    return (r2 * LN2_) * l2;
}

// ---------------------------------------------------------------------------
// Kernel 1: build + solve the 35x35 TPS system per batch (tiny).
// One block of 64 threads per batch; Gaussian elimination w/ partial pivoting.
// ---------------------------------------------------------------------------
__global__ __launch_bounds__(64) void tps_solve_kernel(
    const float* __restrict__ pts_before,  // (N, T, 2)
    const float* __restrict__ pts_after,   // (N, T, 2)
    float* __restrict__ sol)               // (N, 35, 2)
{
    __shared__ float A[NS][NCOLS + 1];     // +1 pad against bank conflicts
    __shared__ float pax[T_], pay[T_], pbx[T_], pby[T_];
    __shared__ float xs[NS][2];
    __shared__ int pivRow;

    const int n = blockIdx.x;
    const int tid = threadIdx.x;

    if (tid < T_) {
        pax[tid] = pts_after[(n * T_ + tid) * 2 + 0] * (2.0f / W_) - 1.0f;
        pay[tid] = pts_after[(n * T_ + tid) * 2 + 1] * (2.0f / H_) - 1.0f;
        pbx[tid] = pts_before[(n * T_ + tid) * 2 + 0] * (2.0f / W_) - 1.0f;
        pby[tid] = pts_before[(n * T_ + tid) * 2 + 1] * (2.0f / H_) - 1.0f;
    }
    __syncthreads();

    // Build augmented [L | V]
    for (int idx = tid; idx < NS * NCOLS; idx += blockDim.x) {
        int i = idx / NCOLS, j = idx % NCOLS;
        float val;
        if (i < T_) {
            if (j < T_) {
                val = (i == j) ? 0.0f : tpsU(pax[i] - pax[j], pay[i] - pay[j]);
            } else if (j == T_)      val = 1.0f;
            else if (j == T_ + 1)    val = pax[i];
            else if (j == T_ + 2)    val = pay[i];
            else if (j == T_ + 3)    val = pbx[i] - pax[i];   // V col x
            else                     val = pby[i] - pay[i];   // V col y
        } else {
            int r = i - T_;
            if (j < T_) val = (r == 0) ? 1.0f : (r == 1 ? pax[j] : pay[j]);
            else        val = 0.0f;
        }
        A[i][j] = val;
    }
    __syncthreads();

    // Gaussian elimination with partial pivoting
    for (int k = 0; k < NS; ++k) {
        if (tid == 0) {
            int p = k; float best = fabsf(A[k][k]);
            for (int i = k + 1; i < NS; ++i) {
                float m = fabsf(A[i][k]);
                if (m > best) { best = m; p = i; }
            }
            pivRow = p;
        }
        __syncthreads();
        int p = pivRow;
        if (p != k && tid < NCOLS) {
            float t = A[k][tid]; A[k][tid] = A[p][tid]; A[p][tid] = t;
        }
        __syncthreads();
        for (int i = k + 1 + tid; i < NS; i += blockDim.x) {
            float f = A[i][k] / A[k][k];
            for (int j = k; j < NCOLS; ++j) A[i][j] -= f * A[k][j];
        }
        __syncthreads();
    }

    // Back substitution (one thread per RHS column)
    if (tid < 2) {
        int c = tid;
        for (int i = NS - 1; i >= 0; --i) {
            float s = A[i][NS + c];
            for (int j = i + 1; j < NS; ++j) s -= A[i][j] * xs[j][c];
            xs[i][c] = s / A[i][i];
        }
    }
    __syncthreads();

    for (int idx = tid; idx < NS * 2; idx += blockDim.x)
        sol[n * NS * 2 + idx] = xs[idx / 2][idx % 2];
}

// ---------------------------------------------------------------------------
// Kernel 2: TPS flow via FP32 WMMA + bilinear sampling.
// Block = 256 threads (8 waves). Each wave handles 32 consecutive pixels as
// two 16-row WMMA tiles. K = 36: 32 TPS kernel values + {1, u, v} affine + pad.
// Per K-step: one ds_load_b128 (4 ctrl-pt coords), one ds_load_b64 (B column
// pair from a zero-padded table), 4 branchless U evals, 2 WMMAs.
// ---------------------------------------------------------------------------
__global__ __launch_bounds__(256) void tps_warp_kernel(
    const float* __restrict__ im,         // (N, C, H, W)
    const float* __restrict__ pts_after,  // (N, T, 2)
    const float* __restrict__ sol,        // (N, 35, 2)
    float* __restrict__ out)              // (N, C, H, W)
{
    __shared__ float s_pap[T_][2];        // interleaved {pax, pay}, 16B-group aligned
    __shared__ float s_solB[16][36];      // B operand table, cols>=2 and row 35 = 0
    __shared__ float s_dxy[8][32][2];     // per-wave flow redistribution

    const int tid = threadIdx.x;
    const int n = blockIdx.y;

    if (tid < T_) {
        s_pap[tid][0] = pts_after[(n * T_ + tid) * 2 + 0] * (2.0f / W_) - 1.0f;
        s_pap[tid][1] = pts_after[(n * T_ + tid) * 2 + 1] * (2.0f / H_) - 1.0f;
    }
    for (int i = tid; i < 16 * 36; i += blockDim.x) {
        int col = i / 36, row = i % 36;
        s_solB[col][row] = (col < 2 && row < NS) ? sol[n * NS * 2 + row * 2 + col] : 0.0f;
    }
    __syncthreads();

    const int lane = tid & 31;
    const int wv = tid >> 5;
    const int halfSel = lane >> 4;       // 0: lanes 0-15, 1: lanes 16-31
    const int mn = lane & 15;            // A: row M;  B: column N

    const int pixBase = blockIdx.x * 256 + wv * 32;   // 32 pixels per wave
    const int h = pixBase >> 9;                        // W_ == 512
    const int w0 = pixBase & (W_ - 1);

    const float vc = h * (2.0f / H_) - 1.0f;
    const float u0 = (w0 + mn) * (2.0f / W_) - 1.0f;        // tile 0 pixel u
    const float u1 = (w0 + 16 + mn) * (2.0f / W_) - 1.0f;   // tile 1 pixel u

    v8f acc0 = {};
    v8f acc1 = {};

    // K = 36 in 9 steps of V_WMMA_F32_16X16X4_F32.
    // A layout: lane M = mn, element e <-> K = 4s + 2*halfSel + e.
    // B layout: lane N = mn, same K mapping.
#pragma unroll
    for (int s = 0; s < 9; ++s) {
        const int k0 = 4 * s + 2 * halfSel;   // always even
        v2f a0, a1;
        if (s < 8) {
            // {pax[k0], pay[k0], pax[k0+1], pay[k0+1]} in one 16B LDS load
            v4f pq = *reinterpret_cast<const v4f*>(&s_pap[k0][0]);
            a0.x = tpsU(u0 - pq.x, vc - pq.y);
            a0.y = tpsU(u0 - pq.z, vc - pq.w);
            a1.x = tpsU(u1 - pq.x, vc - pq.y);
            a1.y = tpsU(u1 - pq.z, vc - pq.w);
        } else {
            // K rows 32..35 = affine basis {1, u, v, 0}
            a0.x = halfSel ? vc   : 1.0f;
            a0.y = halfSel ? 0.0f : u0;
            a1.x = halfSel ? vc   : 1.0f;
            a1.y = halfSel ? 0.0f : u1;
        }
        // Unconditional 8-byte B load from the zero-padded table
        v2f b = *reinterpret_cast<const v2f*>(&s_solB[mn][k0]);

        acc0 = __builtin_amdgcn_wmma_f32_16x16x4_f32(
            false, a0, false, b, (short)0, acc0, false, false);
        acc1 = __builtin_amdgcn_wmma_f32_16x16x4_f32(
            false, a1, false, b, (short)0, acc1, false, false);
    }

    // D element r of lane L: (M = r + 8*halfSel, N = mn). Keep columns 0/1.
    if (mn < 2) {
#pragma unroll
        for (int r = 0; r < 8; ++r) {
            s_dxy[wv][r + 8 * halfSel][mn]      = acc0[r];
            s_dxy[wv][16 + r + 8 * halfSel][mn] = acc1[r];
        }
    }
    __syncthreads();

    // ---- bilinear sampling: one pixel per lane ----
    const int wpix = w0 + lane;
    const float uu = wpix * (2.0f / W_) - 1.0f;
    const float fx = uu + s_dxy[wv][lane][0];
    const float fy = vc + s_dxy[wv][lane][1];

    const float x = (fx + 1.0f) * 0.5f * (W_ - 1);
    const float y = (fy + 1.0f) * 0.5f * (H_ - 1);
    const float x0f = floorf(x), y0f = floorf(y);
    const int ix0 = (int)x0f, iy0 = (int)y0f;
    const int ix1 = ix0 + 1, iy1 = iy0 + 1;
    const float tx = x - x0f, ty = y - y0f;

    const bool vx0 = (ix0 >= 0) & (ix0 < W_);
    const bool vx1 = (ix1 >= 0) & (ix1 < W_);
    const bool vy0 = (iy0 >= 0) & (iy0 < H_);
    const bool vy1 = (iy1 >= 0) & (iy1 < H_);

    // Fold validity into the weights; clamp indices for unconditional loads.
    const float wa = (vx0 && vy0) ? (1.0f - tx) * (1.0f - ty) : 0.0f;  // (x0,y0)
    const float wb = (vx0 && vy1) ? (1.0f - tx) * ty          : 0.0f;  // (x0,y1)
    const float wc = (vx1 && vy0) ? tx * (1.0f - ty)          : 0.0f;  // (x1,y0)
    const float wd = (vx1 && vy1) ? tx * ty                   : 0.0f;  // (x1,y1)

    const int ix0c = min(max(ix0, 0), W_ - 1);
    const int ix1c = min(max(ix1, 0), W_ - 1);
    const int iy0c = min(max(iy0, 0), H_ - 1);
    const int iy1c = min(max(iy1, 0), H_ - 1);
    const int o00 = iy0c * W_ + ix0c;
    const int o01 = iy1c * W_ + ix0c;
    const int o10 = iy0c * W_ + ix1c;
    const int o11 = iy1c * W_ + ix1c;

    const float* imn = im + (size_t)n * C_ * H_ * W_;
#pragma unroll
    for (int c = 0; c < C_; ++c) {
        const float* p = imn + (size_t)c * H_ * W_;
        float acc = p[o00] * wa + p[o01] * wb + p[o10] * wc + p[o11] * wd;
        out[(((size_t)n * C_ + c) * H_ + h) * W_ + wpix] = acc;
    }
}

// ---------------------------------------------------------------------------
extern "C" void kernel_launch(void* const* d_in, const int* in_sizes, int n_in,
                              void* d_out, int out_size, void* d_ws, size_t ws_size,
                              hipStream_t stream) {
    (void)in_sizes; (void)n_in; (void)out_size; (void)ws_size;
    const float* im  = (const float*)d_in[0];
    const float* ptb = (const float*)d_in[1];
    const float* pta = (const float*)d_in[2];
    float* sol = (float*)d_ws;                 // N * 35 * 2 floats
    float* out = (float*)d_out;

    tps_solve_kernel<<<dim3(N_), 64, 0, stream>>>(ptb, pta, sol);
    tps_warp_kernel<<<dim3((H_ * W_) / 256, N_), 256, 0, stream>>>(im, pta, sol, out);
}